// InfoNce_2_11768210391107
// MI455X (gfx1250) — compile-verified
//
#include <hip/hip_runtime.h>

#define NB   4096      // partial blocks for the streaming passes
#define BLK  256       // 8 waves per block (wave32)

typedef __attribute__((ext_vector_type(2))) float v2f;
typedef __attribute__((ext_vector_type(8))) float v8f;

// Sum a float across all 32 lanes of a wave using the f32 matrix unit.
// A (16x4 f32, 2 VGPRs/lane): lanes 0-15 hold {K=0,K=1}, lanes 16-31 {K=2,K=3},
// row M = lane%16. We place v in component 0, zero in component 1, so column
// K=0 carries v[0..15] and column K=2 carries v[16..31]. With B = ones,
// D[m][n] = v[m] + v[m+16]. Lane L then holds 8 row values (M = r or 8+r);
// summing its 8 D VGPRs gives S_low on lanes 0-15 and S_high on lanes 16-31.
// EXEC must be all ones at the call site (call unconditionally, full block).
__device__ __forceinline__ float wave_sum_wmma(float v) {
  v2f a; a[0] = v;    a[1] = 0.0f;
  v2f b; b[0] = 1.0f; b[1] = 1.0f;
  v8f c = {};
  v8f d = __builtin_amdgcn_wmma_f32_16x16x4_f32(
      /*neg_a=*/false, a, /*neg_b=*/false, b,
      /*c_mod=*/(short)0, c, /*reuse_a=*/false, /*reuse_b=*/false);
  float p = d[0] + d[1] + d[2] + d[3] + d[4] + d[5] + d[6] + d[7];
  return __shfl(p, 0, 32) + __shfl(p, 16, 32);
}

__device__ __forceinline__ int wave_sum_i32(int v) {
  #pragma unroll
  for (int m = 16; m > 0; m >>= 1) v += __shfl_xor(v, m, 32);
  return v;
}

// ---------------- Pass 1: partial neg_sum (f32) and n_pos (i32) ----------------
__global__ void pass1_negsum(const float* __restrict__ sims1,
                             const int*   __restrict__ labels,
                             const float* __restrict__ nce,
                             float* __restrict__ partial_neg,
                             int*   __restrict__ partial_cnt,
                             int n) {
  const float inv = 1.0f / nce[0];
  const int tid    = blockIdx.x * BLK + threadIdx.x;
  const int stride = gridDim.x * BLK;
  const int nvec   = n >> 2;

  const float4* __restrict__ s4 = (const float4*)sims1;
  const int4*   __restrict__ l4 = (const int4*)labels;

  float acc = 0.0f;
  int   cnt = 0;
  for (int i = tid; i < nvec; i += stride) {
    float4 x = s4[i];
    int4   l = l4[i];
    float e0 = __expf(x.x * inv), e1 = __expf(x.y * inv);
    float e2 = __expf(x.z * inv), e3 = __expf(x.w * inv);
    acc += (l.x == 0) ? e0 : 0.0f;  cnt += (l.x == 1);
    acc += (l.y == 0) ? e1 : 0.0f;  cnt += (l.y == 1);
    acc += (l.z == 0) ? e2 : 0.0f;  cnt += (l.z == 1);
    acc += (l.w == 0) ? e3 : 0.0f;  cnt += (l.w == 1);
  }
  // scalar tail (N = 2^25 is vec4-exact, but stay general)
  for (int i = (nvec << 2) + tid; i < n; i += stride) {
    float e = __expf(sims1[i] * inv);
    int   l = labels[i];
    acc += (l == 0) ? e : 0.0f;  cnt += (l == 1);
  }

  // wave-level reduce (matrix unit for f32, shuffles for i32)
  float wsum = wave_sum_wmma(acc);
  int   wcnt = wave_sum_i32(cnt);

  __shared__ float sf[BLK / 32];
  __shared__ int   si[BLK / 32];
  const int lane = threadIdx.x & 31;
  const int wave = threadIdx.x >> 5;
  if (lane == 0) { sf[wave] = wsum; si[wave] = wcnt; }
  __syncthreads();
  if (threadIdx.x == 0) {
    float bs = 0.0f; int bc = 0;
    #pragma unroll
    for (int w = 0; w < BLK / 32; ++w) { bs += sf[w]; bc += si[w]; }
    partial_neg[blockIdx.x] = bs;
    partial_cnt[blockIdx.x] = bc;
  }
}

// ---------------- Finalize 1: partials -> scalars ----------------
__global__ void finalize1(const float* __restrict__ partial_neg,
                          const int*   __restrict__ partial_cnt,
                          float* __restrict__ neg_total,
                          int*   __restrict__ cnt_total) {
  const int lane = threadIdx.x;  // 32 threads
  float a = 0.0f; int c = 0;
  for (int i = lane; i < NB; i += 32) { a += partial_neg[i]; c += partial_cnt[i]; }
  float tot = wave_sum_wmma(a);
  int   ct  = wave_sum_i32(c);
  if (lane == 0) { neg_total[0] = tot; cnt_total[0] = ct; }
}

// ---------------- Pass 2: partial sum of positive log-ratios ----------------
__global__ void pass2_pos(const float* __restrict__ sims1,
                          const int*   __restrict__ labels,
                          const float* __restrict__ nce,
                          const float* __restrict__ neg_total,
                          float* __restrict__ partial_pos,
                          int n) {
  const float inv    = 1.0f / nce[0];
  const float negsum = neg_total[0];
  const int tid    = blockIdx.x * BLK + threadIdx.x;
  const int stride = gridDim.x * BLK;
  const int nvec   = n >> 2;

  const float4* __restrict__ s4 = (const float4*)sims1;
  const int4*   __restrict__ l4 = (const int4*)labels;

  float acc = 0.0f;
  for (int i = tid; i < nvec; i += stride) {
    float4 x = s4[i];
    int4   l = l4[i];
    float t0 = x.x * inv, t1 = x.y * inv, t2 = x.z * inv, t3 = x.w * inv;
    // log(e1/denom) = t - log(exp(t) + neg_sum)
    float r0 = t0 - __logf(__expf(t0) + negsum);
    float r1 = t1 - __logf(__expf(t1) + negsum);
    float r2 = t2 - __logf(__expf(t2) + negsum);
    float r3 = t3 - __logf(__expf(t3) + negsum);
    acc += (l.x == 1) ? r0 : 0.0f;
    acc += (l.y == 1) ? r1 : 0.0f;
    acc += (l.z == 1) ? r2 : 0.0f;
    acc += (l.w == 1) ? r3 : 0.0f;
  }
  for (int i = (nvec << 2) + tid; i < n; i += stride) {
    float t = sims1[i] * inv;
    float r = t - __logf(__expf(t) + negsum);
    acc += (labels[i] == 1) ? r : 0.0f;
  }

  float wsum = wave_sum_wmma(acc);
  __shared__ float sf[BLK / 32];
  const int lane = threadIdx.x & 31;
  const int wave = threadIdx.x >> 5;
  if (lane == 0) sf[wave] = wsum;
  __syncthreads();
  if (threadIdx.x == 0) {
    float bs = 0.0f;
    #pragma unroll
    for (int w = 0; w < BLK / 32; ++w) bs += sf[w];
    partial_pos[blockIdx.x] = bs;
  }
}

// ---------------- Finalize 2: write loss ----------------
__global__ void finalize2(const float* __restrict__ partial_pos,
                          const int*   __restrict__ cnt_total,
                          float* __restrict__ out) {
  const int lane = threadIdx.x;  // 32 threads
  float a = 0.0f;
  for (int i = lane; i < NB; i += 32) a += partial_pos[i];
  float tot = wave_sum_wmma(a);
  if (lane == 0) out[0] = -tot / (float)cnt_total[0];
}

extern "C" void kernel_launch(void* const* d_in, const int* in_sizes, int n_in,
                              void* d_out, int out_size, void* d_ws, size_t ws_size,
                              hipStream_t stream) {
  const float* sims1  = (const float*)d_in[0];
  // d_in[1] (sims_0) is dead in the reference -> never touched.
  const int*   labels = (const int*)d_in[2];
  const float* nce    = (const float*)d_in[3];
  float*       out    = (float*)d_out;
  const int n = in_sizes[0];

  // Workspace layout (floats/ints, 64B-aligned regions):
  //   [0]            neg_sum total (f32)
  //   [1]            n_pos total (i32)
  //   [16 .. 16+NB)        partial_neg (f32)
  //   [16+NB .. 16+2NB)    partial_cnt (i32)
  //   [16+2NB .. 16+3NB)   partial_pos (f32)
  float* wsf        = (float*)d_ws;
  int*   wsi        = (int*)d_ws;
  float* neg_total  = wsf;
  int*   cnt_total  = wsi + 1;
  float* p_neg      = wsf + 16;
  int*   p_cnt      = wsi + 16 + NB;
  float* p_pos      = wsf + 16 + 2 * NB;

  pass1_negsum<<<NB, BLK, 0, stream>>>(sims1, labels, nce, p_neg, p_cnt, n);
  finalize1  <<<1, 32, 0, stream>>>(p_neg, p_cnt, neg_total, cnt_total);
  pass2_pos  <<<NB, BLK, 0, stream>>>(sims1, labels, nce, neg_total, p_pos, n);
  finalize2  <<<1, 32, 0, stream>>>(p_pos, cnt_total, out);
}